// GCNonLocal_65300682768404
// MI455X (gfx1250) — compile-verified
//
#include <hip/hip_runtime.h>
#include <math.h>

// ---------------------------------------------------------------------------
// GCNet-style global-context block, MI455X (gfx1250, wave32).
// x: [B=16, C=256, H*W=S=16384] fp32.
// Memory-bound (~1 GB traffic -> ~45us @ 23.3 TB/s). Tiny GEMM chain done with
// V_WMMA_F32_16X16X4_F32.
// ---------------------------------------------------------------------------

#define B_ 16
#define C_ 256
#define S_ 16384           // 128*128
#define CR_ 32             // C/8
#define EPS_ 1e-5f

typedef float v2f __attribute__((ext_vector_type(2)));
typedef float v8f __attribute__((ext_vector_type(8)));

// ---------------------------------------------------------------------------
// Kernel 1: scores[b,s] = sum_c x[b,c,s] * w_reduce[c]
// grid: 16 batches * 16 s-chunks = 256 blocks, 256 threads, 4 s per thread
// (float4 / global_load_b128, coalesced across s; w_reduce[c] is uniform ->
// scalar loads). One full pass over x.
// ---------------------------------------------------------------------------
__global__ __launch_bounds__(256) void k_scores(const float* __restrict__ x,
                                                const float* __restrict__ wr,
                                                float* __restrict__ scores) {
    const int b  = blockIdx.x >> 4;
    const int s0 = ((blockIdx.x & 15) << 10) + (threadIdx.x << 2);
    const float* xb = x + (size_t)b * C_ * S_;
    float ax = 0.f, ay = 0.f, az = 0.f, aw = 0.f;
#pragma unroll 4
    for (int c = 0; c < C_; ++c) {
        const float  w = wr[c];                       // uniform -> SGPR
        const float4 v = *(const float4*)(xb + (size_t)c * S_ + s0);
        ax = fmaf(v.x, w, ax);
        ay = fmaf(v.y, w, ay);
        az = fmaf(v.z, w, az);
        aw = fmaf(v.w, w, aw);
    }
    float4 r; r.x = ax; r.y = ay; r.z = az; r.w = aw;
    *(float4*)(scores + b * S_ + s0) = r;
}

// ---------------------------------------------------------------------------
// Kernel 2: per-batch softmax over S, attn written in-place over scores.
// Also zeroes ctx[b,:] for the atomic accumulation in kernel 3.
// grid: 16 blocks (one per batch), 256 threads, 64 elems/thread.
// ---------------------------------------------------------------------------
__global__ __launch_bounds__(256) void k_softmax(float* __restrict__ sc,
                                                 float* __restrict__ ctx) {
    __shared__ float red[256];
    const int b   = blockIdx.x;
    const int tid = threadIdx.x;
    float* p = sc + b * S_;

    // max
    float m = -3.402823466e+38f;
#pragma unroll 8
    for (int i = 0; i < 64; ++i) m = fmaxf(m, p[tid + (i << 8)]);
    red[tid] = m; __syncthreads();
    for (int o = 128; o > 0; o >>= 1) {
        if (tid < o) red[tid] = fmaxf(red[tid], red[tid + o]);
        __syncthreads();
    }
    m = red[0]; __syncthreads();

    // sum of exp
    float z = 0.f;
#pragma unroll 8
    for (int i = 0; i < 64; ++i) z += __expf(p[tid + (i << 8)] - m);
    red[tid] = z; __syncthreads();
    for (int o = 128; o > 0; o >>= 1) {
        if (tid < o) red[tid] += red[tid + o];
        __syncthreads();
    }
    const float inv_z = 1.0f / red[0];

    // attn in place (each thread owns its indices)
#pragma unroll 8
    for (int i = 0; i < 64; ++i) {
        const int s = tid + (i << 8);
        p[s] = __expf(p[s] - m) * inv_z;
    }

    ctx[b * C_ + tid] = 0.f;   // 256 ctx entries per batch, one per thread
}

// ---------------------------------------------------------------------------
// Kernel 3: ctx[b,c] += sum_{s in chunk} x[b,c,s] * attn[b,s]
// grid: 16 batches * 8 s-chunks (2048 each) = 128 blocks, 256 threads.
// attn chunk staged in LDS once, x streamed once (second full pass over x).
// wave32 shuffle reduction; per-wave partials in LDS; one atomicAdd per (blk,c).
// ---------------------------------------------------------------------------
__global__ __launch_bounds__(256) void k_ctx(const float* __restrict__ x,
                                             const float* __restrict__ attn,
                                             float* __restrict__ ctx) {
    __shared__ float attnS[2048];
    __shared__ float part[C_ * 8];          // [c][wave]
    const int b    = blockIdx.x >> 3;
    const int s0   = (blockIdx.x & 7) << 11;
    const int tid  = threadIdx.x;
    const int wave = tid >> 5;
    const int lane = tid & 31;

#pragma unroll
    for (int i = 0; i < 8; ++i)
        attnS[tid + (i << 8)] = attn[b * S_ + s0 + tid + (i << 8)];
    __syncthreads();

    const float* xb = x + (size_t)b * C_ * S_ + s0;
    for (int c = 0; c < C_; ++c) {
        const float* xc = xb + (size_t)c * S_;
        float acc = 0.f;
#pragma unroll
        for (int i = 0; i < 8; ++i)
            acc = fmaf(xc[tid + (i << 8)], attnS[tid + (i << 8)], acc);
        // wave32 reduce
        for (int o = 16; o > 0; o >>= 1) acc += __shfl_down(acc, o, 32);
        if (lane == 0) part[(c << 3) + wave] = acc;
    }
    __syncthreads();
    // tid < 256 == blockDim: each thread finalizes one c
    float t = 0.f;
#pragma unroll
    for (int i = 0; i < 8; ++i) t += part[(tid << 3) + i];
    atomicAdd(&ctx[b * C_ + tid], t);
}

// ---------------------------------------------------------------------------
// Kernel 4 (single block, 512 threads = 16 waves): tiny GEMM chain with WMMA.
//   y[16,32]   = ctx[16,256] x w_squeeze^T          (V_WMMA_F32_16X16X4_F32)
//   y_norm     = gamma*((y-y)*rsqrt(eps)) + beta    (InstanceNorm on 1x1)
//   r          = relu(y_norm)
//   att[16,256]= r[16,32] x w_expand^T              (V_WMMA_F32_16X16X4_F32)
//   scale      = 1 + att
// A layout (32-bit 16x4): lanes 0-15 -> K=0(x),1(y); lanes 16-31 -> K=2,3.
// C/D layout: VGPR i -> M = i + (lane<16?0:8), N = lane&15.
// EXEC is all-ones for every wave executing WMMA.
// ---------------------------------------------------------------------------
__global__ __launch_bounds__(512) void k_gemm_chain(const float* __restrict__ ctx,
                                                    const float* __restrict__ w_sq,
                                                    const float* __restrict__ gamma,
                                                    const float* __restrict__ beta,
                                                    const float* __restrict__ w_ex,
                                                    float* __restrict__ scale) {
    __shared__ float yS[B_][CR_];
    __shared__ float rS[B_][CR_];
    const int tid  = threadIdx.x;
    const int wave = tid >> 5;
    const int lane = tid & 31;
    const int m    = lane & 15;          // A-row / output N-column within tile
    const int koff = (lane >> 4) << 1;   // 0 for lanes 0-15, 2 for lanes 16-31
    const int mrow = (lane >> 4) << 3;   // C/D: +8 row offset for upper lanes

    // ---- GEMM1: y = ctx (M=b,K=c) x B (K=c, N=o), B[k][n] = w_squeeze[n][k]
    if (wave < 2) {
        const int ntile = wave;          // o columns [16*ntile, 16*ntile+15]
        const int n     = (ntile << 4) + m;
        v8f acc = {};
#pragma unroll 8
        for (int k0 = 0; k0 < C_; k0 += 4) {
            v2f a, bm;
            a.x  = ctx[m * C_ + k0 + koff];
            a.y  = ctx[m * C_ + k0 + koff + 1];
            bm.x = w_sq[n * C_ + k0 + koff];
            bm.y = w_sq[n * C_ + k0 + koff + 1];
            acc = __builtin_amdgcn_wmma_f32_16x16x4_f32(
                false, a, false, bm, (short)0, acc, false, false);
        }
#pragma unroll
        for (int i = 0; i < 8; ++i)
            yS[i + mrow][(ntile << 4) + m] = acc[i];
    }
    __syncthreads();

    // ---- InstanceNorm(1x1) + ReLU, computed faithfully: mean=y, var=0
    {
        const int bb = tid >> 5;         // 0..15
        const int oo = tid & 31;         // 0..31
        const float yv = yS[bb][oo];
        const float yn = (yv - yv) * rsqrtf(EPS_);      // exactly 0
        rS[bb][oo] = fmaxf(fmaf(gamma[oo], yn, beta[oo]), 0.f);
    }
    __syncthreads();

    // ---- GEMM2: att = r (M=b,K=o) x B (K=o, N=c), B[k][n] = w_expand[n][k]
    {
        const int n = (wave << 4) + m;   // output channel c for this lane
        v8f acc = {};
#pragma unroll
        for (int k0 = 0; k0 < CR_; k0 += 4) {
            v2f a, bm;
            a.x  = rS[m][k0 + koff];
            a.y  = rS[m][k0 + koff + 1];
            bm.x = w_ex[n * CR_ + k0 + koff];
            bm.y = w_ex[n * CR_ + k0 + koff + 1];
            acc = __builtin_amdgcn_wmma_f32_16x16x4_f32(
                false, a, false, bm, (short)0, acc, false, false);
        }
#pragma unroll
        for (int i = 0; i < 8; ++i)
            scale[(i + mrow) * C_ + n] = 1.0f + acc[i];   // fold residual
    }
}

// ---------------------------------------------------------------------------
// Kernel 5: out[b,c,s] = scale[b,c] * x[b,c,s]   (== att*x + x)
// grid: 65536 blocks * 256 threads * float4. scale index uniform per block ->
// scalar load. Third pass over x + the 256 MB store.
// ---------------------------------------------------------------------------
__global__ __launch_bounds__(256) void k_apply(const float* __restrict__ x,
                                               const float* __restrict__ scale,
                                               float* __restrict__ out) {
    const size_t idx = ((size_t)blockIdx.x * 256 + threadIdx.x) << 2;
    const float  s   = scale[blockIdx.x >> 4];     // 16 blocks per (b,c)
    float4 v = *(const float4*)(x + idx);
    v.x *= s; v.y *= s; v.z *= s; v.w *= s;
    *(float4*)(out + idx) = v;
}

// ---------------------------------------------------------------------------
extern "C" void kernel_launch(void* const* d_in, const int* in_sizes, int n_in,
                              void* d_out, int out_size, void* d_ws, size_t ws_size,
                              hipStream_t stream) {
    const float* x     = (const float*)d_in[0];   // [16,256,128,128]
    const float* w_red = (const float*)d_in[1];   // [256]
    const float* w_sq  = (const float*)d_in[2];   // [32,256]
    const float* gamma = (const float*)d_in[3];   // [32]
    const float* beta  = (const float*)d_in[4];   // [32]
    const float* w_ex  = (const float*)d_in[5];   // [256,32]
    float*       out   = (float*)d_out;

    // workspace layout (floats): scores/attn [16*16384], ctx [16*256], scale [16*256]
    float* scores = (float*)d_ws;
    float* ctx    = scores + B_ * S_;
    float* scale  = ctx + B_ * C_;

    k_scores <<<B_ * 16, 256, 0, stream>>>(x, w_red, scores);
    k_softmax<<<B_,      256, 0, stream>>>(scores, ctx);
    k_ctx    <<<B_ * 8,  256, 0, stream>>>(x, scores, ctx);
    k_gemm_chain<<<1,    512, 0, stream>>>(ctx, w_sq, gamma, beta, w_ex, scale);
    k_apply  <<<(B_ * C_ * S_) / (256 * 4), 256, 0, stream>>>(x, scale, out);
}